// StructuredPruningIF_18193481466348
// MI455X (gfx1250) — compile-verified
//
#include <hip/hip_runtime.h>

typedef float float4_ __attribute__((ext_vector_type(4)));

#define TSTEPS 4
#define BLK 256

// asm (not builtin) so the wait is also a compiler memory barrier: LDS reads
// must not be scheduled above the s_wait_asynccnt that guarantees arrival.
#define WAIT_ASYNC(n) asm volatile("s_wait_asynccnt %0" ::"n"(n) : "memory")

// Issue one async global->LDS 128-bit load per timestep plane for this thread.
// The LDS byte address is taken from the *actual* shared-array pointer: on
// amdgcn the low 32 bits of a generic LDS address are the wave-relative LDS
// offset, and the ptrtoint capture makes the shared array escape, so the
// compiler must assume this asm writes it (keeps the ds_load consumption live).
__device__ __forceinline__ void stage_chunk(const float* __restrict__ x,
                                            long long planeN, long long baseElem,
                                            int tid, float4_ (*buf)[BLK]) {
#pragma unroll
  for (int p = 0; p < TSTEPS; ++p) {
    unsigned long long g =
        (unsigned long long)(x + (long long)p * planeN + baseElem) +
        (unsigned long long)tid * 16ull;
    unsigned l = (unsigned)(unsigned long long)&buf[p][tid];
    asm volatile("global_load_async_to_lds_b128 %0, %1, off"
                 :: "v"(l), "v"(g)
                 : "memory");
  }
}

__global__ void __launch_bounds__(BLK) if_spike_async(
    const float* __restrict__ x, const float* __restrict__ threshp,
    float* __restrict__ out, long long planeN, int nchunks) {
  __shared__ float4_ sh[2][TSTEPS][BLK];  // 32 KB double buffer

  const int tid = threadIdx.x;
  long long c = blockIdx.x;
  if (c >= (long long)nchunks) return;

  const float thre = threshp[0];
  const long long stride = gridDim.x;

  // Preload first chunk into buffer 0.
  stage_chunk(x, planeN, c * (long long)(BLK * 4), tid, sh[0]);

  int cur = 0;
  for (; c < (long long)nchunks; c += stride) {
    const long long nxt = c + stride;
    const bool have_next = (nxt < (long long)nchunks);
    if (have_next) {
      // Prefetch next chunk into the other buffer, then retire the older 4
      // async loads (ASYNCcnt loads complete in order).
      stage_chunk(x, planeN, nxt * (long long)(BLK * 4), tid, sh[cur ^ 1]);
      WAIT_ASYNC(4);
    } else {
      WAIT_ASYNC(0);
    }

    // Consume this thread's staged data from LDS (ds_load_b128 x4).
    float4_ a[TSTEPS];
#pragma unroll
    for (int p = 0; p < TSTEPS; ++p) a[p] = sh[cur][p][tid];

    // Branch-free integrate-and-fire scan over T, vectorized over 4 lanes.
    float4_ s[TSTEPS];
#pragma unroll
    for (int j = 0; j < 4; ++j) {
      float mem = 0.5f * thre;
#pragma unroll
      for (int t = 0; t < TSTEPS; ++t) {
        mem += a[t][j];
        float sp = (mem >= thre) ? thre : 0.0f;
        mem -= sp;
        s[t][j] = sp;
      }
    }

    // Streaming non-temporal 128-bit stores straight to HBM.
    const long long baseElem = c * (long long)(BLK * 4) + (long long)tid * 4;
#pragma unroll
    for (int p = 0; p < TSTEPS; ++p) {
      __builtin_nontemporal_store(
          s[p], (float4_*)(out + (long long)p * planeN + baseElem));
    }

    cur ^= 1;
  }
}

// Scalar tail for sizes not divisible by 4*BLK elements per plane.
__global__ void __launch_bounds__(BLK) if_spike_tail(
    const float* __restrict__ x, const float* __restrict__ threshp,
    float* __restrict__ out, long long planeN, long long start) {
  long long i = start + (long long)blockIdx.x * BLK + threadIdx.x;
  if (i >= planeN) return;
  const float thre = threshp[0];
  float mem = 0.5f * thre;
#pragma unroll
  for (int t = 0; t < TSTEPS; ++t) {
    mem += x[(long long)t * planeN + i];
    float sp = (mem >= thre) ? thre : 0.0f;
    mem -= sp;
    out[(long long)t * planeN + i] = sp;
  }
}

extern "C" void kernel_launch(void* const* d_in, const int* in_sizes, int n_in,
                              void* d_out, int out_size, void* d_ws,
                              size_t ws_size, hipStream_t stream) {
  const float* x = (const float*)d_in[0];
  const float* th = (const float*)d_in[1];
  float* out = (float*)d_out;

  const long long total = (long long)in_sizes[0];   // T*B*C*H*W
  const long long planeN = total / TSTEPS;          // elements per t-plane
  const long long nvec = planeN / 4;                // float4s per plane
  const int nchunks = (int)(nvec / BLK);            // 256-float4 chunks
  const long long covered = (long long)nchunks * BLK * 4;

  if (nchunks > 0) {
    int grid = nchunks < 4096 ? nchunks : 4096;  // >=2 chunks/block -> pipeline
    if_spike_async<<<grid, BLK, 0, stream>>>(x, th, out, planeN, nchunks);
  }
  if (covered < planeN) {
    long long rem = planeN - covered;
    int tb = (int)((rem + BLK - 1) / BLK);
    if_spike_tail<<<tb, BLK, 0, stream>>>(x, th, out, planeN, covered);
  }
}